// quickGCN_6717328851470
// MI455X (gfx1250) — compile-verified
//
#include <hip/hip_runtime.h>
#include <cstdint>
#include <cstddef>

#define NODES 100000
#define EDGES 1600000
#define FIN   128
#define HID   128
#define NCLS  64

typedef __attribute__((ext_vector_type(2))) float v2f;
typedef __attribute__((ext_vector_type(8))) float v8f;

// ---------------------------------------------------------------- utilities

__global__ void k_zero(float* __restrict__ p, long n) {
  long i = (long)blockIdx.x * blockDim.x + threadIdx.x;
  long stride = (long)gridDim.x * blockDim.x;
  for (; i < n; i += stride) p[i] = 0.0f;
}

__global__ void k_degree(const long long* __restrict__ dst,
                         float* __restrict__ deg) {
  int e = blockIdx.x * blockDim.x + threadIdx.x;
  if (e < EDGES) unsafeAtomicAdd(&deg[(int)dst[e]], 1.0f);
}

__global__ void k_recip(const float* __restrict__ deg,
                        float* __restrict__ dinv) {
  int i = blockIdx.x * blockDim.x + threadIdx.x;
  if (i < NODES) dinv[i] = 1.0f / fmaxf(deg[i], 1.0f);
}

// One wave (32 lanes) per edge; each lane moves 4 consecutive features with a
// float4 load and 4 hardware fp32 atomics. x/h (51 MB) live in the 192 MB L2,
// so the random per-edge traffic stays on-chip.
__global__ void k_scatter(const float* __restrict__ feat,
                          const long long* __restrict__ src,
                          const long long* __restrict__ dst,
                          float* __restrict__ acc) {
  long g = (long)blockIdx.x * blockDim.x + threadIdx.x;
  int e = (int)(g >> 5);
  int l = (int)(g & 31);
  if (e >= EDGES) return;
  int s = (int)src[e];
  int d = (int)dst[e];
  const float4 v = *(const float4*)(feat + (size_t)s * FIN + l * 4);
  float* o = acc + (size_t)d * FIN + l * 4;
  unsafeAtomicAdd(o + 0, v.x);
  unsafeAtomicAdd(o + 1, v.y);
  unsafeAtomicAdd(o + 2, v.z);
  unsafeAtomicAdd(o + 3, v.w);
}

// ------------------------------------------------- WMMA f32 16x16x4 helpers
// A frag (16x4 f32): lane L holds A[M=L&15][K = 2*(L>>4) + j], j=0,1 -> v2f
// B frag (4x16 f32): lane L holds B[K = 2*(L>>4) + j][N=L&15]
// D/C   (16x16 f32): VGPR v, lane L -> (M = v + 8*(L>>4), N = L&15)
template <int NT>
__device__ __forceinline__ void mm_phase(v8f acc[NT],
                                         const float* __restrict__ arow,
                                         float ascale,
                                         const float* __restrict__ sWp,
                                         int ldw, int half, int m) {
#pragma unroll 4
  for (int kk = 0; kk < 128; kk += 4) {
    const int k = kk + 2 * half;
    v2f a = *(const v2f*)(arow + k);   // 8B-aligned (k is even)
    a.x *= ascale;
    a.y *= ascale;
#pragma unroll
    for (int t = 0; t < NT; ++t) {
      v2f b;
      b.x = sWp[(size_t)k * ldw + t * 16 + m];
      b.y = sWp[(size_t)(k + 1) * ldw + t * 16 + m];
      acc[t] = __builtin_amdgcn_wmma_f32_16x16x4_f32(
          false, a, false, b, (short)0, acc[t], false, false);
    }
  }
}

// -------------------------------------------------------------- layer 1
// h = relu( (sum1 * dinv) @ W1_l + x @ W1_r + b1 )
// One wave computes a 16-row x 128-col output block (8 accumulator tiles).
// Weights staged through 64 KB static LDS, one 128x128 half at a time.
__global__ void k_sage1(const float* __restrict__ sum1,
                        const float* __restrict__ x,
                        const float* __restrict__ dinv,
                        const float* __restrict__ Wl,
                        const float* __restrict__ Wr,
                        const float* __restrict__ bias,
                        float* __restrict__ h, int ntiles) {
  __shared__ float sW[HID * HID];  // 64 KB
  const int tid = threadIdx.x;
  const int wave = (blockIdx.x * blockDim.x + tid) >> 5;
  const int lane = tid & 31;
  const int m = lane & 15;
  const int half = lane >> 4;
  const bool active = (wave < ntiles);
  const int rowBase = wave * 16;
  const int row = rowBase + m;

  v8f acc[8] = {};

  // phase 1: (sum1 * dinv) @ W1_l
  for (int i = tid * 4; i < HID * HID; i += blockDim.x * 4)
    *(float4*)(sW + i) = *(const float4*)(Wl + i);
  __syncthreads();
  if (active) {
    const float ds = dinv[row];
    mm_phase<8>(acc, sum1 + (size_t)row * FIN, ds, sW, HID, half, m);
  }
  __syncthreads();

  // phase 2: x @ W1_r
  for (int i = tid * 4; i < HID * HID; i += blockDim.x * 4)
    *(float4*)(sW + i) = *(const float4*)(Wr + i);
  __syncthreads();
  if (active) {
    mm_phase<8>(acc, x + (size_t)row * FIN, 1.0f, sW, HID, half, m);
#pragma unroll
    for (int t = 0; t < 8; ++t) {
      const float bc = bias[t * 16 + m];
#pragma unroll
      for (int v = 0; v < 8; ++v) {
        const int r = rowBase + v + 8 * half;
        h[(size_t)r * HID + t * 16 + m] = fmaxf(acc[t][v] + bc, 0.0f);
      }
    }
  }
}

// -------------------------------------------------------------- layer 2
// out = log_softmax( (sum2 * dinv) @ W2_l + h @ W2_r + b2 )
// One wave computes a full 16-row x 64-col block, so log-softmax is done
// entirely in registers with shfl_xor reductions over 16-lane halves.
__global__ void k_sage2(const float* __restrict__ sum2,
                        const float* __restrict__ hin,
                        const float* __restrict__ dinv,
                        const float* __restrict__ Wl,
                        const float* __restrict__ Wr,
                        const float* __restrict__ bias,
                        float* __restrict__ out, int ntiles) {
  __shared__ float sW[2 * HID * NCLS];  // 64 KB: [W2_l ; W2_r]
  const int tid = threadIdx.x;
  for (int i = tid * 4; i < HID * NCLS; i += blockDim.x * 4) {
    *(float4*)(sW + i) = *(const float4*)(Wl + i);
    *(float4*)(sW + HID * NCLS + i) = *(const float4*)(Wr + i);
  }
  __syncthreads();

  const int wave = (blockIdx.x * blockDim.x + tid) >> 5;
  if (wave >= ntiles) return;
  const int lane = tid & 31;
  const int m = lane & 15;
  const int half = lane >> 4;
  const int rowBase = wave * 16;
  const int row = rowBase + m;
  const float ds = dinv[row];

  v8f acc[4] = {};
  mm_phase<4>(acc, sum2 + (size_t)row * HID, ds, sW, NCLS, half, m);
  mm_phase<4>(acc, hin + (size_t)row * HID, 1.0f, sW + HID * NCLS, NCLS, half, m);

  float bv[4];
#pragma unroll
  for (int t = 0; t < 4; ++t) bv[t] = bias[t * 16 + m];

#pragma unroll
  for (int v = 0; v < 8; ++v) {
    float lv[4];
    float mx = -3.402823466e38f;
#pragma unroll
    for (int t = 0; t < 4; ++t) {
      lv[t] = acc[t][v] + bv[t];
      mx = fmaxf(mx, lv[t]);
    }
    // row lives across one 16-lane half; xor masks < 16 stay in the half
#pragma unroll
    for (int mask = 1; mask < 16; mask <<= 1)
      mx = fmaxf(mx, __shfl_xor(mx, mask, 32));
    float se = 0.0f;
#pragma unroll
    for (int t = 0; t < 4; ++t) se += expf(lv[t] - mx);
#pragma unroll
    for (int mask = 1; mask < 16; mask <<= 1)
      se += __shfl_xor(se, mask, 32);
    const float lse = mx + logf(se);
    const int r = rowBase + v + 8 * half;
#pragma unroll
    for (int t = 0; t < 4; ++t)
      out[(size_t)r * NCLS + t * 16 + m] = lv[t] - lse;
  }
}

// ---------------------------------------------------------------- launcher

extern "C" void kernel_launch(void* const* d_in, const int* in_sizes, int n_in,
                              void* d_out, int out_size, void* d_ws,
                              size_t ws_size, hipStream_t stream) {
  const float* x = (const float*)d_in[0];
  const long long* ei = (const long long*)d_in[1];  // [2, E] int64
  const float* W1l = (const float*)d_in[2];
  const float* W1r = (const float*)d_in[3];
  const float* b1 = (const float*)d_in[4];
  const float* W2l = (const float*)d_in[5];
  const float* W2r = (const float*)d_in[6];
  const float* b2 = (const float*)d_in[7];
  float* out = (float*)d_out;

  float* ws = (float*)d_ws;
  float* sum1 = ws;                            // N*128
  float* sum2 = sum1 + (size_t)NODES * FIN;    // N*128
  float* deg = sum2 + (size_t)NODES * FIN;     // N
  float* dinv = deg + NODES;                   // N
  float* h = dinv + NODES;                     // N*128

  const long long* srcI = ei;
  const long long* dstI = ei + EDGES;

  // zero sum1, sum2, deg (must re-zero every call: atomics accumulate)
  const long zn = (long)NODES * (2 * FIN) + NODES;
  k_zero<<<2048, 256, 0, stream>>>(ws, zn);

  k_degree<<<(EDGES + 255) / 256, 256, 0, stream>>>(dstI, deg);
  k_scatter<<<(EDGES * 32 + 255) / 256, 256, 0, stream>>>(x, srcI, dstI, sum1);
  k_recip<<<(NODES + 255) / 256, 256, 0, stream>>>(deg, dinv);

  const int ntiles = NODES / 16;                     // 6250 (exact)
  const int blocks = (ntiles * 32 + 255) / 256;      // 8 waves/block
  k_sage1<<<blocks, 256, 0, stream>>>(sum1, x, dinv, W1l, W1r, b1, h, ntiles);

  k_scatter<<<(EDGES * 32 + 255) / 256, 256, 0, stream>>>(h, srcI, dstI, sum2);
  k_sage2<<<blocks, 256, 0, stream>>>(sum2, h, dinv, W2l, W2r, b2, out, ntiles);
}